// VAE_32341103738993
// MI455X (gfx1250) — compile-verified
//
#include <hip/hip_runtime.h>
#include <hip/hip_bf16.h>
#include <math.h>
#include <stdint.h>

// VAE forward for MI455X (gfx1250, wave32, WMMA f32 16x16x4).
#define IO    64
#define LAT   32
#define HID   60
#define TNUM  8
#define WAVES 8
#define BLOCK (WAVES * 32)

typedef float v2f __attribute__((ext_vector_type(2)));
typedef float v8f __attribute__((ext_vector_type(8)));

// ---------------- LDS layout (float offsets) ----------------
// Weight packs are pre-swizzled into WMMA B-fragment order:
// pack[(kk*NT + nt)*64 + lane*2 + j] = B[k = 4*kk + 2*(lane>>4) + j][n = nt*16 + (lane&15)]
#define KS_ENC1 16
#define NT_ENC1 4
#define KS_MS   15
#define NT_MS   2
#define KS_DEC1 8
#define NT_DEC1 4
#define KS_DEC2 15
#define NT_DEC2 4

#define P_ENC1 0
#define P_MEAN (P_ENC1 + KS_ENC1*NT_ENC1*64)   //  4096
#define P_STD  (P_MEAN + KS_MS*NT_MS*64)       // +1920
#define P_DEC1 (P_STD  + KS_MS*NT_MS*64)       // +1920
#define P_DEC2 (P_DEC1 + KS_DEC1*NT_DEC1*64)   // +2048
#define P_END  (P_DEC2 + KS_DEC2*NT_DEC2*64)   // +3840
#define B_ENC1 (P_END)          // 64 (padded)
#define B_MEAN (B_ENC1 + 64)    // 32
#define B_STD  (B_MEAN + 32)    // 32
#define B_DEC1 (B_STD  + 32)    // 64 (padded)
#define B_DEC2 (B_DEC1 + 64)    // 64
#define SH_END (B_DEC2 + 64)

// per-wave scratch (floats)
#define W_BUF 0      // 16x64 : h1, later traj-out staging
#define W_MU  1024   // 16x32
#define W_SD  1536   // 16x32
#define W_H3  2048   // 16x64
#define W_PP  3072   // 8x16 p values
#define W_PT  3200   // 32 partial sums
#define W_DEN 3232   // 16 denominators
#define WAVE_F 3264

#define LDS_TOTAL (SH_END + WAVES * WAVE_F)   // 40192 floats = 157 KB

// LDS ops from one wave are in-order; this waits for returns + is a compiler fence.
#define LDS_SYNC() asm volatile("s_wait_dscnt 0x0" ::: "memory")

static __device__ __forceinline__ v8f wmma4(v2f a, v2f b, v8f c) {
  return __builtin_amdgcn_wmma_f32_16x16x4_f32(false, a, false, b, (short)0, c,
                                               false, false);
}

// ---------------- threefry2x32 (JAX key(1) => (0,1)) ----------------
static __device__ __forceinline__ uint32_t rotl32(uint32_t x, uint32_t r) {
  return (x << r) | (x >> (32u - r));
}
static __device__ __forceinline__ void tf4(uint32_t& x0, uint32_t& x1,
                                           int r0, int r1, int r2, int r3) {
  x0 += x1; x1 = rotl32(x1, r0); x1 ^= x0;
  x0 += x1; x1 = rotl32(x1, r1); x1 ^= x0;
  x0 += x1; x1 = rotl32(x1, r2); x1 ^= x0;
  x0 += x1; x1 = rotl32(x1, r3); x1 ^= x0;
}
static __device__ __forceinline__ void threefry2x32(uint32_t c0, uint32_t c1,
                                                    uint32_t& o0, uint32_t& o1) {
  const uint32_t K0 = 0u, K1 = 1u, K2 = 0x1BD11BDAu ^ K0 ^ K1;  // 0x1BD11BDB
  uint32_t x0 = c0 + K0, x1 = c1 + K1;
  tf4(x0, x1, 13, 15, 26, 6);  x0 += K1; x1 += K2 + 1u;
  tf4(x0, x1, 17, 29, 16, 24); x0 += K2; x1 += K0 + 2u;
  tf4(x0, x1, 13, 15, 26, 6);  x0 += K0; x1 += K1 + 3u;
  tf4(x0, x1, 17, 29, 16, 24); x0 += K1; x1 += K2 + 4u;
  tf4(x0, x1, 13, 15, 26, 6);  x0 += K2; x1 += K0 + 5u;
  o0 = x0; o1 = x1;
}
static __device__ __forceinline__ float erfinv_f(float u) {
  float w = -__logf(fmaxf((1.0f - u) * (1.0f + u), 1e-38f));
  float p;
  if (w < 5.0f) {
    w -= 2.5f;
    p = 2.81022636e-08f;
    p = fmaf(p, w, 3.43273939e-07f);
    p = fmaf(p, w, -3.5233877e-06f);
    p = fmaf(p, w, -4.39150654e-06f);
    p = fmaf(p, w, 0.00021858087f);
    p = fmaf(p, w, -0.00125372503f);
    p = fmaf(p, w, -0.00417768164f);
    p = fmaf(p, w, 0.246640727f);
    p = fmaf(p, w, 1.50140941f);
  } else {
    w = __fsqrt_rn(w) - 3.0f;
    p = -0.000200214257f;
    p = fmaf(p, w, 0.000100950558f);
    p = fmaf(p, w, 0.00134934322f);
    p = fmaf(p, w, -0.00367342844f);
    p = fmaf(p, w, 0.00573950773f);
    p = fmaf(p, w, -0.0076224613f);
    p = fmaf(p, w, 0.00943887047f);
    p = fmaf(p, w, 1.00167406f);
    p = fmaf(p, w, 2.83297682f);
  }
  return p * u;
}
// eps[t][n][l], flat f = (t*N+n)*32+l, JAX pairs (f, f+half)
static __device__ __forceinline__ float eps_normal(uint32_t f, uint32_t half) {
  uint32_t c0 = (f >= half) ? (f - half) : f;
  uint32_t o0, o1;
  threefry2x32(c0, c0 + half, o0, o1);
  uint32_t bits = (f >= half) ? o1 : o0;
  float u01 = __uint_as_float((bits >> 9) | 0x3f800000u) - 1.0f;
  float u = fmaf(u01, 2.0f, -1.0f);
  u = fminf(fmaxf(u, -0.99999994f), 0.99999994f);
  return 1.41421356f * erfinv_f(u);
}

// Stage weight matrix W[Nreal][K] (row-major, B = W^T) into fragment pack.
static __device__ __forceinline__ void stage_pack(float* lds, int off,
                                                  const float* W, int Nreal,
                                                  int K, int KS, int NT, int tid) {
  int total = KS * NT * 64;
  for (int i = tid; i < total; i += BLOCK) {
    int q = i & 63;
    int s = i >> 6;
    int nt = s % NT, kk = s / NT;
    int l = q >> 1, j = q & 1;
    int n = nt * 16 + (l & 15);
    int k = 4 * kk + 2 * (l >> 4) + j;
    lds[off + i] = (n < Nreal) ? W[n * K + k] : 0.0f;
  }
}
static __device__ __forceinline__ void stage_bias(float* lds, int off,
                                                  const float* b, int nreal,
                                                  int npad, int tid) {
  for (int i = tid; i < npad; i += BLOCK) lds[off + i] = (i < nreal) ? b[i] : 0.0f;
}

__global__ __launch_bounds__(BLOCK, 1) void VAE_32341103738993_kernel(
    const float* __restrict__ x, const float* __restrict__ e1w,
    const float* __restrict__ e1b, const float* __restrict__ mw,
    const float* __restrict__ mb, const float* __restrict__ sw,
    const float* __restrict__ sb, const float* __restrict__ d1w,
    const float* __restrict__ d1b, const float* __restrict__ d2w,
    const float* __restrict__ d2b, float* __restrict__ out, int N) {
  __shared__ float lds[LDS_TOTAL];
  const int tid = threadIdx.x;

  stage_pack(lds, P_ENC1, e1w, HID, IO, KS_ENC1, NT_ENC1, tid);
  stage_pack(lds, P_MEAN, mw, LAT, HID, KS_MS, NT_MS, tid);
  stage_pack(lds, P_STD, sw, LAT, HID, KS_MS, NT_MS, tid);
  stage_pack(lds, P_DEC1, d1w, HID, LAT, KS_DEC1, NT_DEC1, tid);
  stage_pack(lds, P_DEC2, d2w, IO, HID, KS_DEC2, NT_DEC2, tid);
  stage_bias(lds, B_ENC1, e1b, HID, 64, tid);
  stage_bias(lds, B_MEAN, mb, LAT, 32, tid);
  stage_bias(lds, B_STD, sb, LAT, 32, tid);
  stage_bias(lds, B_DEC1, d1b, HID, 64, tid);
  stage_bias(lds, B_DEC2, d2b, IO, 64, tid);
  __syncthreads();

  const int wave = tid >> 5, lane = tid & 31;
  const int lr = lane & 15, hi = lane >> 4;
  float* wbuf = lds + SH_END + wave * WAVE_F;
  const int base = (blockIdx.x * WAVES + wave) * 16;
  if (base >= N) return;

  float* traj = out;
  float* mu_o = out + (size_t)TNUM * 16 * N * 4;
  float* lv_o = mu_o + (size_t)N * LAT;
  float* pr_o = lv_o + (size_t)N * LAT;

  // ---------------- enc1: h1 = relu(x @ W^T + b), 16x60 (pad 64) ----------
  {
    v8f acc[NT_ENC1];
#pragma unroll
    for (int nt = 0; nt < NT_ENC1; ++nt) {
      float bc = lds[B_ENC1 + nt * 16 + lr];
      acc[nt] = (v8f){bc, bc, bc, bc, bc, bc, bc, bc};
    }
    const float* xrow = x + (size_t)(base + lr) * IO + 2 * hi;
#pragma unroll
    for (int kk = 0; kk < KS_ENC1; ++kk) {
      v2f a = *(const v2f*)(xrow + 4 * kk);
#pragma unroll
      for (int nt = 0; nt < NT_ENC1; ++nt) {
        v2f b = *(const v2f*)&lds[P_ENC1 + (kk * NT_ENC1 + nt) * 64 + lane * 2];
        acc[nt] = wmma4(a, b, acc[nt]);
      }
    }
#pragma unroll
    for (int nt = 0; nt < NT_ENC1; ++nt)
#pragma unroll
      for (int i = 0; i < 8; ++i)
        wbuf[W_BUF + (i + 8 * hi) * 64 + nt * 16 + lr] = fmaxf(acc[nt][i], 0.0f);
    LDS_SYNC();
  }

  // ---------------- mean / logvar heads: [16x60] @ [60x32] ----------------
  {
    v8f am[NT_MS], as[NT_MS];
#pragma unroll
    for (int nt = 0; nt < NT_MS; ++nt) {
      float bm = lds[B_MEAN + nt * 16 + lr], bs = lds[B_STD + nt * 16 + lr];
      am[nt] = (v8f){bm, bm, bm, bm, bm, bm, bm, bm};
      as[nt] = (v8f){bs, bs, bs, bs, bs, bs, bs, bs};
    }
#pragma unroll
    for (int kk = 0; kk < KS_MS; ++kk) {
      v2f a = *(const v2f*)&wbuf[W_BUF + lr * 64 + 4 * kk + 2 * hi];
#pragma unroll
      for (int nt = 0; nt < NT_MS; ++nt) {
        v2f bmv = *(const v2f*)&lds[P_MEAN + (kk * NT_MS + nt) * 64 + lane * 2];
        v2f bsv = *(const v2f*)&lds[P_STD + (kk * NT_MS + nt) * 64 + lane * 2];
        am[nt] = wmma4(a, bmv, am[nt]);
        as[nt] = wmma4(a, bsv, as[nt]);
      }
    }
#pragma unroll
    for (int nt = 0; nt < NT_MS; ++nt)
#pragma unroll
      for (int i = 0; i < 8; ++i) {
        int m = i + 8 * hi, c = nt * 16 + lr;
        float mu = am[nt][i], lv = as[nt][i];
        mu_o[(size_t)(base + m) * LAT + c] = mu;
        lv_o[(size_t)(base + m) * LAT + c] = lv;
        wbuf[W_MU + m * LAT + c] = mu;
        wbuf[W_SD + m * LAT + c] = __expf(0.5f * lv);
      }
    LDS_SYNC();
  }

  // ---------------- trajectories ------------------------------------------
  const uint32_t half = (uint32_t)TNUM * (uint32_t)N * 16u;
  for (int t = 0; t < TNUM; ++t) {
    // dec1: z = mu + eps*std (built in A-fragment layout), h3 = relu(z@W^T+b)
    float ss = 0.0f;
    v8f a1[NT_DEC1];
#pragma unroll
    for (int nt = 0; nt < NT_DEC1; ++nt) {
      float bc = lds[B_DEC1 + nt * 16 + lr];
      a1[nt] = (v8f){bc, bc, bc, bc, bc, bc, bc, bc};
    }
#pragma unroll
    for (int kk = 0; kk < KS_DEC1; ++kk) {
      int k0 = 4 * kk + 2 * hi;
      v2f mu2 = *(const v2f*)&wbuf[W_MU + lr * LAT + k0];
      v2f sd2 = *(const v2f*)&wbuf[W_SD + lr * LAT + k0];
      uint32_t f0 = ((uint32_t)t * (uint32_t)N + (uint32_t)(base + lr)) * 32u +
                    (uint32_t)k0;
      float e0 = eps_normal(f0, half);
      float e1 = eps_normal(f0 + 1u, half);
      ss = fmaf(e0, e0, ss);
      ss = fmaf(e1, e1, ss);
      v2f a;
      a.x = fmaf(e0, sd2.x, mu2.x);
      a.y = fmaf(e1, sd2.y, mu2.y);
#pragma unroll
      for (int nt = 0; nt < NT_DEC1; ++nt) {
        v2f b = *(const v2f*)&lds[P_DEC1 + (kk * NT_DEC1 + nt) * 64 + lane * 2];
        a1[nt] = wmma4(a, b, a1[nt]);
      }
    }
#pragma unroll
    for (int nt = 0; nt < NT_DEC1; ++nt)
#pragma unroll
      for (int i = 0; i < 8; ++i)
        wbuf[W_H3 + (i + 8 * hi) * 64 + nt * 16 + lr] = fmaxf(a1[nt][i], 0.0f);

    // pdf: combine the two lane-halves' partial sum of eps^2 per row
    wbuf[W_PT + lane] = ss;
    LDS_SYNC();
    if (hi == 0) {
      float tot = wbuf[W_PT + lr] + wbuf[W_PT + 16 + lr];
      float lp = -0.5f * (32.0f * 1.8378770664093453f + tot);
      wbuf[W_PP + t * 16 + lr] = __expf(__expf(lp));  // faithful: exp(pdf)
    }
    LDS_SYNC();

    // dec2: out = h3 @ W^T + b  (16x64)
    v8f a2[NT_DEC2];
#pragma unroll
    for (int nt = 0; nt < NT_DEC2; ++nt) {
      float bc = lds[B_DEC2 + nt * 16 + lr];
      a2[nt] = (v8f){bc, bc, bc, bc, bc, bc, bc, bc};
    }
#pragma unroll
    for (int kk = 0; kk < KS_DEC2; ++kk) {
      v2f a = *(const v2f*)&wbuf[W_H3 + lr * 64 + 4 * kk + 2 * hi];
#pragma unroll
      for (int nt = 0; nt < NT_DEC2; ++nt) {
        v2f b = *(const v2f*)&lds[P_DEC2 + (kk * NT_DEC2 + nt) * 64 + lane * 2];
        a2[nt] = wmma4(a, b, a2[nt]);
      }
    }
    // stage to LDS, then coalesced 16B stores into [T,H,N,4] transposed layout
#pragma unroll
    for (int nt = 0; nt < NT_DEC2; ++nt)
#pragma unroll
      for (int i = 0; i < 8; ++i)
        wbuf[W_BUF + (i + 8 * hi) * 64 + nt * 16 + lr] = a2[nt][i];
    LDS_SYNC();
#pragma unroll
    for (int it = 0; it < 8; ++it) {
      int q = it * 32 + lane;
      int r = q & 15, h = q >> 4;
      float4 v = *(const float4*)&wbuf[W_BUF + r * 64 + h * 4];
      *(float4*)&traj[(((size_t)t * 16 + h) * N + base + r) * 4] = v;
    }
    LDS_SYNC();
  }

  // ---------------- prob_list: p_t / sum_t p_t ----------------------------
  if (hi == 0) {
    float d = 0.0f;
#pragma unroll
    for (int tt = 0; tt < TNUM; ++tt) d += wbuf[W_PP + tt * 16 + lr];
    wbuf[W_DEN + lr] = d;
  }
  LDS_SYNC();
  const int g = base >> 4, ng = N >> 4;
#pragma unroll
  for (int pass = 0; pass < 4; ++pass) {
    int tt = pass * 2 + hi;
    float v = wbuf[W_PP + tt * 16 + lr] / wbuf[W_DEN + lr];
    pr_o[((size_t)tt * ng + g) * 16 + lr] = v;
  }
}

extern "C" void kernel_launch(void* const* d_in, const int* in_sizes, int n_in,
                              void* d_out, int out_size, void* d_ws,
                              size_t ws_size, hipStream_t stream) {
  const float* x   = (const float*)d_in[0];
  const float* e1w = (const float*)d_in[2];
  const float* e1b = (const float*)d_in[3];
  const float* mw  = (const float*)d_in[4];
  const float* mb  = (const float*)d_in[5];
  const float* sw  = (const float*)d_in[6];
  const float* sb  = (const float*)d_in[7];
  const float* d1w = (const float*)d_in[8];
  const float* d1b = (const float*)d_in[9];
  const float* d2w = (const float*)d_in[10];
  const float* d2b = (const float*)d_in[11];
  float* out = (float*)d_out;
  const int N = in_sizes[0] / IO;  // 65536 rows
  const int grid = (N + 16 * WAVES - 1) / (16 * WAVES);
  VAE_32341103738993_kernel<<<grid, BLOCK, 0, stream>>>(
      x, e1w, e1b, mw, mb, sw, sb, d1w, d1b, d2w, d2b, out, N);
}